// EvolveGCN_30124900614685
// MI455X (gfx1250) — compile-verified
//
#include <hip/hip_runtime.h>
#include <math.h>

#define D 64
#define HDIM 4096     // D*D, GRU hidden size
#define G3H 12288     // 3*HDIM

typedef __attribute__((ext_vector_type(2))) float v2f;
typedef __attribute__((ext_vector_type(8))) float v8f;

__device__ __forceinline__ float wave_reduce32(float v) {
    #pragma unroll
    for (int off = 16; off > 0; off >>= 1) v += __shfl_down(v, off, 32);
    return v;
}

// y1 = W@v1 + b, y2 = W@v2 + b. W: [rows, 4096]. One wave per row.
// Fuses both GRU cells so W (201 MB) streams from HBM exactly once.
__global__ void matvec2_kernel(const float* __restrict__ W,
                               const float* __restrict__ v1,
                               const float* __restrict__ v2,
                               const float* __restrict__ bias,
                               float* __restrict__ y1,
                               float* __restrict__ y2,
                               int rows) {
    const int wavesPerBlk = blockDim.x >> 5;
    const int row  = blockIdx.x * wavesPerBlk + (threadIdx.x >> 5);
    const int lane = threadIdx.x & 31;
    if (row >= rows) return;
    const float4* Wr = (const float4*)(W + (size_t)row * HDIM);
    const float4* a4 = (const float4*)v1;
    const float4* b4 = (const float4*)v2;
    float acc1 = 0.f, acc2 = 0.f;
    #pragma unroll 4
    for (int i = lane; i < HDIM / 4; i += 32) {
        float4 w = Wr[i];
        float4 a = a4[i];
        float4 b = b4[i];
        acc1 += w.x*a.x + w.y*a.y + w.z*a.z + w.w*a.w;
        acc2 += w.x*b.x + w.y*b.y + w.z*b.z + w.w*b.w;
    }
    acc1 = wave_reduce32(acc1);
    acc2 = wave_reduce32(acc2);
    if (lane == 0) {
        float bb = bias[row];
        y1[row] = acc1 + bb;
        y2[row] = acc2 + bb;
    }
}

// GRU elementwise combine: wout = (1-z)*n + z*h
__global__ void gru_combine_kernel(const float* __restrict__ gx,
                                   const float* __restrict__ gh,
                                   const float* __restrict__ hcur,
                                   float* __restrict__ wout) {
    int i = blockIdx.x * blockDim.x + threadIdx.x;
    if (i >= HDIM) return;
    float r = 1.f / (1.f + __expf(-(gx[i]          + gh[i])));
    float z = 1.f / (1.f + __expf(-(gx[HDIM + i]   + gh[HDIM + i])));
    float n = tanhf(gx[2*HDIM + i] + r * gh[2*HDIM + i]);
    wout[i] = (1.f - z) * n + z * hcur[i];
}

__global__ void degree_kernel(const int* __restrict__ src, const int* __restrict__ dst,
                              float* __restrict__ deg_out, float* __restrict__ deg_in,
                              int nE) {
    int e = blockIdx.x * blockDim.x + threadIdx.x;
    if (e >= nE) return;
    atomicAdd(&deg_out[src[e]], 1.0f);
    atomicAdd(&deg_in [dst[e]], 1.0f);
}

// in-place: deg -> rsqrt(max(deg, 1))
__global__ void rnorm_kernel(float* __restrict__ a, float* __restrict__ b, int n) {
    int i = blockIdx.x * blockDim.x + threadIdx.x;
    if (i >= n) return;
    a[i] = rsqrtf(fmaxf(a[i], 1.0f));
    b[i] = rsqrtf(fmaxf(b[i], 1.0f));
}

// h[node, :] = (x[node, :] * rnorm[node]) @ W  via V_WMMA_F32_16X16X4_F32.
// Block = 128 threads (4 waves). Block owns 16 nodes; wave w owns output cols
// [16w, 16w+16). W (16KB) and the 16 scaled rows staged in LDS.
__global__ void __launch_bounds__(128)
node_xform_wmma_kernel(const float* __restrict__ x, const float* __restrict__ rnorm,
                       const float* __restrict__ w, float* __restrict__ h, int n) {
    __shared__ float s_w[D * D];   // 16 KB
    __shared__ float s_x[16 * D];  // 4 KB (pre-scaled rows)
    const int tid  = threadIdx.x;
    const int lane = tid & 31;
    const int wave = tid >> 5;
    const int nodeBase = blockIdx.x * 16;

    // stage weights (coalesced float4)
    {
        const float4* w4 = (const float4*)w;
        float4* s4 = (float4*)s_w;
        #pragma unroll
        for (int i = tid; i < D * D / 4; i += 128) s4[i] = w4[i];
    }
    // stage 16 node rows, scaled by rsqrt(deg_out)
    {
        #pragma unroll
        for (int i = tid; i < 16 * D / 4; i += 128) {
            int r = i >> 4;                       // 16 float4 per row
            int node = nodeBase + r;
            int nc = node < n ? node : (n - 1);
            float s = rnorm[nc];
            float4 v = ((const float4*)(x + (size_t)nc * D))[i & 15];
            v.x *= s; v.y *= s; v.z *= s; v.w *= s;
            ((float4*)s_x)[i] = v;
        }
    }
    __syncthreads();

    // 16x4 f32 A layout: lane L -> M = L&15, K-offset = (L<16)?0:2 (2 VGPRs)
    // 4x16 f32 B layout: lane L -> N = L&15, same K split (2 VGPRs)
    const int m    = lane & 15;
    const int koff = (lane < 16) ? 0 : 2;
    const int nc   = wave * 16 + (lane & 15);     // output column
    v8f acc = {};
    #pragma unroll
    for (int k0 = 0; k0 < D; k0 += 4) {
        v2f a = *(const v2f*)(s_x + m * D + k0 + koff);   // 8B aligned ds_load
        v2f b;
        b.x = s_w[(k0 + koff)     * D + nc];
        b.y = s_w[(k0 + koff + 1) * D + nc];
        acc = __builtin_amdgcn_wmma_f32_16x16x4_f32(
                  false, a, false, b, (short)0, acc, false, false);
    }
    // 16x16 f32 C/D layout: lanes 0-15 -> M = v, lanes 16-31 -> M = v+8; N = lane&15
    const int mbase = (lane < 16) ? 0 : 8;
    #pragma unroll
    for (int v = 0; v < 8; ++v) {
        int node = nodeBase + mbase + v;
        if (node < n) h[(size_t)node * D + nc] = acc[v];
    }
}

// agg[dst[e], :] += h[src[e], :]   (float4 per thread, 16 threads per edge)
__global__ void scatter_kernel(const float* __restrict__ h,
                               const int* __restrict__ src, const int* __restrict__ dst,
                               float* __restrict__ agg, int nE) {
    long long t = (long long)blockIdx.x * blockDim.x + threadIdx.x;
    int e = (int)(t >> 4);
    if (e >= nE) return;
    int f = ((int)t & 15) * 4;
    int s = src[e], d = dst[e];
    float4 v = *(const float4*)(h + (size_t)s * D + f);
    float* p = agg + (size_t)d * D + f;
    atomicAdd(p + 0, v.x);
    atomicAdd(p + 1, v.y);
    atomicAdd(p + 2, v.z);
    atomicAdd(p + 3, v.w);
}

// out = [relu]( agg * rsqrt(deg_in)[node] + bias[feat] )
__global__ void finalize_kernel(const float* __restrict__ agg,
                                const float* __restrict__ rnorm_in,
                                const float* __restrict__ bias,
                                float* __restrict__ out,
                                int total, int do_relu) {
    int i = blockIdx.x * blockDim.x + threadIdx.x;
    if (i >= total) return;
    int node = i >> 6;
    int f    = i & 63;
    float v = agg[i] * rnorm_in[node] + bias[f];
    out[i] = do_relu ? fmaxf(v, 0.0f) : v;
}

extern "C" void kernel_launch(void* const* d_in, const int* in_sizes, int n_in,
                              void* d_out, int out_size, void* d_ws, size_t ws_size,
                              hipStream_t stream) {
    const float* x0    = (const float*)d_in[0];
    const int*   src   = (const int*)  d_in[1];
    const int*   dst   = (const int*)  d_in[2];
    const float* gc1_w = (const float*)d_in[3];
    const float* gc1_b = (const float*)d_in[4];
    const float* gc2_w = (const float*)d_in[5];
    const float* gc2_b = (const float*)d_in[6];
    const float* gc1_h = (const float*)d_in[7];
    const float* gc2_h = (const float*)d_in[8];
    const float* w_ih  = (const float*)d_in[9];
    const float* w_hh  = (const float*)d_in[10];
    const float* b_ih  = (const float*)d_in[11];
    const float* b_hh  = (const float*)d_in[12];

    const int nNodes = in_sizes[0] / D;
    const int nEdges = in_sizes[1];
    const int totalF = nNodes * D;

    // workspace carve-up (floats, 256B-aligned chunks)
    float* ws = (float*)d_ws;
    size_t off = 0;
    auto carve = [&](size_t n) { float* p = ws + off; off += (n + 63) & ~(size_t)63; return p; };
    float* w1   = carve(HDIM);
    float* w2   = carve(HDIM);
    float* gx1  = carve(G3H);
    float* gx2  = carve(G3H);
    float* gh1  = carve(G3H);
    float* gh2  = carve(G3H);
    float* rno  = carve(nNodes);              // becomes rsqrt(deg_out)
    float* rni  = carve(nNodes);              // becomes rsqrt(deg_in)
    float* bufA = carve((size_t)nNodes * D);  // h1 -> relu(x2)
    float* bufB = carve((size_t)nNodes * D);  // agg1 -> h2
    float* outF = (float*)d_out;              // agg2 accumulates here

    // zero accumulators (graph-capture-safe memset nodes)
    hipMemsetAsync(rno,  0, sizeof(float) * nNodes, stream);
    hipMemsetAsync(rni,  0, sizeof(float) * nNodes, stream);
    hipMemsetAsync(bufB, 0, sizeof(float) * (size_t)nNodes * D, stream);
    hipMemsetAsync(outF, 0, sizeof(float) * (size_t)out_size, stream);

    // --- GRU weight evolution (each weight matrix streamed exactly once) ---
    const int mvBlocks = (G3H + 7) / 8;  // 8 waves/block, 1 wave per row
    matvec2_kernel<<<mvBlocks, 256, 0, stream>>>(w_ih, gc1_h, gc2_h, b_ih, gx1, gx2, G3H);
    matvec2_kernel<<<mvBlocks, 256, 0, stream>>>(w_hh, gc1_w, gc2_w, b_hh, gh1, gh2, G3H);
    gru_combine_kernel<<<(HDIM + 255) / 256, 256, 0, stream>>>(gx1, gh1, gc1_w, w1);
    gru_combine_kernel<<<(HDIM + 255) / 256, 256, 0, stream>>>(gx2, gh2, gc2_w, w2);

    // --- degrees -> rsqrt norms ---
    degree_kernel<<<(nEdges + 255) / 256, 256, 0, stream>>>(src, dst, rno, rni, nEdges);
    rnorm_kernel<<<(nNodes + 255) / 256, 256, 0, stream>>>(rno, rni, nNodes);

    // --- layer 1: transform (WMMA), scatter, bias+relu ---
    const int xfBlocks = (nNodes + 15) / 16;
    const long long scThreads = (long long)nEdges * 16;
    const int scBlocks = (int)((scThreads + 255) / 256);
    node_xform_wmma_kernel<<<xfBlocks, 128, 0, stream>>>(x0, rno, w1, bufA, nNodes);
    scatter_kernel<<<scBlocks, 256, 0, stream>>>(bufA, src, dst, bufB, nEdges);
    finalize_kernel<<<(totalF + 255) / 256, 256, 0, stream>>>(bufB, rni, gc1_b, bufA, totalF, 1);

    // --- layer 2: transform (WMMA), scatter into d_out, bias ---
    node_xform_wmma_kernel<<<xfBlocks, 128, 0, stream>>>(bufA, rno, w2, bufB, nNodes);
    scatter_kernel<<<scBlocks, 256, 0, stream>>>(bufB, src, dst, outF, nEdges);
    finalize_kernel<<<(totalF + 255) / 256, 256, 0, stream>>>(outF, rni, gc2_b, outF, totalF, 0);
}